// GResNet_86964497809766
// MI455X (gfx1250) — compile-verified
//
#include <hip/hip_runtime.h>

typedef __attribute__((ext_vector_type(2))) float v2f;
typedef __attribute__((ext_vector_type(8))) float v8f;

#define NFEAT 128
#define NHID  128
#define NCLASS 64

// ---------------- utility kernels ----------------

__global__ void k_zero_f4(float4* __restrict__ p, unsigned n4) {
    unsigned i = blockIdx.x * blockDim.x + threadIdx.x;
    if (i < n4) p[i] = make_float4(0.f, 0.f, 0.f, 0.f);
}

__global__ void k_zero_i32(int* __restrict__ p, unsigned n) {
    unsigned i = blockIdx.x * blockDim.x + threadIdx.x;
    if (i < n) p[i] = 0;
}

__global__ void k_degrees(const int* __restrict__ src, const int* __restrict__ dst,
                          int* __restrict__ outdeg, int* __restrict__ indeg, unsigned E) {
    unsigned i = blockIdx.x * blockDim.x + threadIdx.x;
    if (i < E) {
        atomicAdd(&outdeg[src[i]], 1);
        atomicAdd(&indeg[dst[i]], 1);
    }
}

__global__ void k_norms(const int* __restrict__ outdeg, const int* __restrict__ indeg,
                        float* __restrict__ ns, float* __restrict__ nd, unsigned N) {
    unsigned i = blockIdx.x * blockDim.x + threadIdx.x;
    if (i < N) {
        ns[i] = rsqrtf(fmaxf((float)outdeg[i], 1.0f));
        nd[i] = rsqrtf(fmaxf((float)indeg[i], 1.0f));
    }
}

// ---- f32 WMMA GEMM: Y[M,Nout] = X[M,K] @ W[K,Nout], compile-time row scale -------
// One wave computes one 16x16 tile; 8 waves/block share the A tile across N tiles.

template<int K, int Nout, bool SCALE>
__global__ void k_gemm_wmma(const float* __restrict__ X, const float* __restrict__ W,
                            float* __restrict__ Y, const float* __restrict__ rowScale,
                            int M) {
    const int lane = threadIdx.x & 31;
    const int tile = blockIdx.x * (blockDim.x >> 5) + (threadIdx.x >> 5);
    constexpr int TN = Nout >> 4;
    const int mt = tile / TN;                       // TN is 8 or 4 -> shift
    const int nt = tile & (TN - 1);
    const int mbase = mt << 4;
    if (mbase >= M) return;

    const int lmod  = lane & 15;
    const int lhalf = lane >> 4;                    // 0: K=k,k+1   1: K=k+2,k+3

    // 32-bit offsets: all buffers < 2^31 elements
    unsigned aoff = (unsigned)(mbase + lmod) * K + (lhalf << 1);  // X[row][kk]
    unsigned boff = (unsigned)(lhalf << 1) * Nout + ((nt << 4) + lmod);

    v8f acc = {};
#if __has_builtin(__builtin_amdgcn_wmma_f32_16x16x4_f32)
    #pragma unroll 8
    for (int k = 0; k < K; k += 4) {
        v2f a = *(const v2f*)(X + aoff);            // X[row][kk..kk+1] contiguous
        v2f b;
        b.x = W[boff];                              // W[kk][col]
        b.y = W[boff + Nout];                       // W[kk+1][col]
        acc = __builtin_amdgcn_wmma_f32_16x16x4_f32(
            false, a, false, b, (short)0, acc, false, false);
        aoff += 4;
        boff += 4u * Nout;
    }
#else
    for (int k = 0; k < K; k += 4) {
        v2f a = *(const v2f*)(X + aoff);
        acc[0] += a.x * W[boff] + a.y * W[boff + Nout];
        aoff += 4;
        boff += 4u * Nout;
    }
#endif

    // C/D layout: VGPR r -> row = mbase + r + 8*lhalf, col = nbase + (lane&15)
    const unsigned colu = (unsigned)((nt << 4) + lmod);
    const unsigned yrow = (unsigned)(mbase + (lhalf << 3));

    if (SCALE) {
        float sc[8];
        #pragma unroll
        for (int r = 0; r < 8; ++r) sc[r] = rowScale[yrow + r];
        #pragma unroll
        for (int r = 0; r < 8; ++r)
            Y[(yrow + r) * (unsigned)Nout + colu] = acc[r] * sc[r];
    } else {
        #pragma unroll
        for (int r = 0; r < 8; ++r)
            Y[(yrow + r) * (unsigned)Nout + colu] = acc[r];
    }
}

// ---- edge gather + atomic scatter-add for both GC and residual streams -----------
// F==128: one wave == one edge; src/dst are wave-uniform -> readfirstlane (scalar).

template<int F>
__global__ void k_edge_agg(const float* __restrict__ U, const float* __restrict__ V,
                           const int* __restrict__ src, const int* __restrict__ dst,
                           float* __restrict__ Uagg, float* __restrict__ Vagg,
                           unsigned total) {
    unsigned gid = blockIdx.x * blockDim.x + threadIdx.x;
    if (gid >= total) return;
    constexpr int CH = F >> 2;                      // float4 chunks per row
    constexpr int CSH = (F == 128) ? 5 : 4;
    unsigned e = gid >> CSH;
    unsigned c = (gid & (CH - 1)) << 2;

    int s = src[e];
    int d = dst[e];
    if (F == 128) {                                 // whole wave shares one edge
        s = __builtin_amdgcn_readfirstlane(s);
        d = __builtin_amdgcn_readfirstlane(d);
    }
    const unsigned so = (unsigned)s * F + c;        // < 12.8M, 32-bit
    const unsigned dofs = (unsigned)d * F + c;

    const float4 u = *(const float4*)(U + so);
    const float4 v = *(const float4*)(V + so);
    float* up = Uagg + dofs;
    float* vp = Vagg + dofs;
    atomicAdd(up + 0, u.x); atomicAdd(up + 1, u.y);
    atomicAdd(up + 2, u.z); atomicAdd(up + 3, u.w);
    atomicAdd(vp + 0, v.x); atomicAdd(vp + 1, v.y);
    atomicAdd(vp + 2, v.z); atomicAdd(vp + 3, v.w);
}

// ---- combine: out = maybe_relu(Uagg*norm_d + bias + Vagg) ------------------------

template<int F, bool RELU>
__global__ void k_combine(const float* __restrict__ Uagg, const float* __restrict__ Vagg,
                          const float* __restrict__ nd, const float* __restrict__ bias,
                          float* __restrict__ out, unsigned total) {
    unsigned gid = blockIdx.x * blockDim.x + threadIdx.x;
    if (gid >= total) return;
    constexpr int FSH = (F == 128) ? 7 : 6;
    const unsigned f = gid & (F - 1);
    const unsigned i = gid >> FSH;
    float v = fmaf(Uagg[gid], nd[i], bias[f]) + Vagg[gid];
    out[gid] = RELU ? fmaxf(v, 0.0f) : v;
}

// ---------------- host orchestration ----------------

extern "C" void kernel_launch(void* const* d_in, const int* in_sizes, int n_in,
                              void* d_out, int out_size, void* d_ws, size_t ws_size,
                              hipStream_t stream) {
    const float* raw_x = (const float*)d_in[0];
    const int*   src   = (const int*)d_in[1];
    const int*   dst   = (const int*)d_in[2];
    const float* gc_w0 = (const float*)d_in[3];
    const float* gc_b0 = (const float*)d_in[4];
    const float* gc_w1 = (const float*)d_in[5];
    const float* gc_b1 = (const float*)d_in[6];
    const float* gc_w2 = (const float*)d_in[7];
    const float* gc_b2 = (const float*)d_in[8];
    const float* rw0   = (const float*)d_in[9];
    const float* rw1   = (const float*)d_in[10];
    const float* rw2   = (const float*)d_in[11];

    const int N = in_sizes[0] / NFEAT;     // 100000
    const int E = in_sizes[1];             // 1600000
    const size_t NB = (size_t)N * NFEAT;

    float* P0 = (float*)d_ws;
    float* P1 = P0 + NB;
    float* P2 = P1 + NB;
    float* P3 = P2 + NB;
    float* ns = P3 + NB;
    float* nd = ns + N;
    int*  odeg = (int*)(nd + N);
    int*  ideg = odeg + N;

    const int TB = 256;
    auto blocks = [](long long n, int tb) { return (int)((n + tb - 1) / tb); };

    // degrees + norms
    k_zero_i32<<<blocks(2LL * N, TB), TB, 0, stream>>>(odeg, 2u * N);
    k_degrees<<<blocks(E, TB), TB, 0, stream>>>(src, dst, odeg, ideg, (unsigned)E);
    k_norms<<<blocks(N, TB), TB, 0, stream>>>(odeg, ideg, ns, nd, (unsigned)N);

    auto zero = [&](float* p, size_t n) {
        k_zero_f4<<<blocks((long long)(n >> 2), TB), TB, 0, stream>>>(
            (float4*)p, (unsigned)(n >> 2));
    };

    const int mtiles = N / 16;             // 6250, N % 16 == 0
    auto gemm128s = [&](const float* X, const float* W, float* Y) {
        k_gemm_wmma<128, 128, true><<<blocks((long long)mtiles * 8, 8), TB, 0, stream>>>(
            X, W, Y, ns, N);
    };
    auto gemm128 = [&](const float* X, const float* W, float* Y) {
        k_gemm_wmma<128, 128, false><<<blocks((long long)mtiles * 8, 8), TB, 0, stream>>>(
            X, W, Y, nullptr, N);
    };
    auto gemm64s = [&](const float* X, const float* W, float* Y) {
        k_gemm_wmma<128, 64, true><<<blocks((long long)mtiles * 4, 8), TB, 0, stream>>>(
            X, W, Y, ns, N);
    };
    auto gemm64 = [&](const float* X, const float* W, float* Y) {
        k_gemm_wmma<128, 64, false><<<blocks((long long)mtiles * 4, 8), TB, 0, stream>>>(
            X, W, Y, nullptr, N);
    };

    auto edge128 = [&](const float* U, const float* V, float* Ua, float* Va) {
        unsigned total = (unsigned)E << 5;
        k_edge_agg<128><<<blocks(total, TB), TB, 0, stream>>>(U, V, src, dst, Ua, Va, total);
    };
    auto edge64 = [&](const float* U, const float* V, float* Ua, float* Va) {
        unsigned total = (unsigned)E << 4;
        k_edge_agg<64><<<blocks(total, TB), TB, 0, stream>>>(U, V, src, dst, Ua, Va, total);
    };

    // ---- Layer 0: x = raw_x ----
    gemm128s(raw_x, gc_w0, P1);                     // U = (x@W0)*norm_s
    gemm128 (raw_x, rw0,   P2);                     // V = x@R0
    zero(P0, NB); zero(P3, NB);
    edge128(P1, P2, P0, P3);                        // Uagg=P0, Vagg=P3
    {
        unsigned total = (unsigned)N * 128u;
        k_combine<128, true><<<blocks(total, TB), TB, 0, stream>>>(
            P0, P3, nd, gc_b0, P1, total);          // x1 = P1
    }

    // ---- Layer 1: x = P1 ----
    gemm128s(P1, gc_w1, P2);                        // U = P2
    gemm128 (P1, rw1,   P0);                        // V = P0
    zero(P3, NB); zero(P1, NB);
    edge128(P2, P0, P3, P1);                        // Uagg=P3, Vagg=P1
    {
        unsigned total = (unsigned)N * 128u;
        k_combine<128, true><<<blocks(total, TB), TB, 0, stream>>>(
            P3, P1, nd, gc_b1, P0, total);          // x2 = P0
    }

    // ---- Layer 2: x = P0, F_out = 64 ----
    gemm64s(P0, gc_w2, P1);                         // U = P1
    gemm64 (P0, rw2,   P2);                         // V = P2
    const size_t NB2 = (size_t)N * NCLASS;
    zero(P3, NB2); zero(P0, NB2);
    edge64(P1, P2, P3, P0);                         // Uagg=P3, Vagg=P0
    {
        unsigned total = (unsigned)N * 64u;
        k_combine<64, false><<<blocks(total, TB), TB, 0, stream>>>(
            P3, P0, nd, gc_b2, (float*)d_out, total);
    }
}